// GICPModel_65910568124475
// MI455X (gfx1250) — compile-verified
//
#include <hip/hip_runtime.h>

// ext-vector types for WMMA operands (f32 16x16x4: A/B = 2 VGPRs, C/D = 8 VGPRs, wave32)
typedef float v2f __attribute__((ext_vector_type(2)));
typedef float v8f __attribute__((ext_vector_type(8)));

// One chained K16 @ CVEC accumulation for a 16-point batch:
//   D += sum_k A_k(16x4) @ B_k(4x16),  B_k[kk][n] = C_n[kk][k]
// ra = this lane's C row (0 or 2 by lane half), rb = row (1 or 3).
__device__ __forceinline__ v8f kron_sandwich(const v2f* Ak, float4 ra, float4 rb, v8f c) {
    v2f b;
    b.x = ra.x; b.y = rb.x;
    c = __builtin_amdgcn_wmma_f32_16x16x4_f32(false, Ak[0], false, b, (short)0, c, false, false);
    b.x = ra.y; b.y = rb.y;
    c = __builtin_amdgcn_wmma_f32_16x16x4_f32(false, Ak[1], false, b, (short)0, c, false, false);
    b.x = ra.z; b.y = rb.z;
    c = __builtin_amdgcn_wmma_f32_16x16x4_f32(false, Ak[2], false, b, (short)0, c, false, false);
    b.x = ra.w; b.y = rb.w;
    c = __builtin_amdgcn_wmma_f32_16x16x4_f32(false, Ak[3], false, b, (short)0, c, false, false);
    return c;
}

__global__ __launch_bounds__(256) void gicp_main(
    const float* __restrict__ T,         // 4x4 row-major
    const float* __restrict__ src,       // N x 4
    const float* __restrict__ tar,       // M x 4
    const float* __restrict__ covs_src,  // N x 4 x 4
    const float* __restrict__ covs_tar,  // M x 4 x 4
    const int*   __restrict__ idx,       // N
    float* __restrict__ partial,         // one per block
    int N, int numBatches)
{
    const int tid           = threadIdx.x;
    const int lane          = tid & 31;
    const int waveInBlock   = tid >> 5;
    const int wavesPerBlock = blockDim.x >> 5;
    const int waveId        = blockIdx.x * wavesPerBlock + waveInBlock;
    const int totalWaves    = gridDim.x * wavesPerBlock;

    // Uniform copy of T for the residual transform (static indices -> SGPRs).
    float Tm[16];
#pragma unroll
    for (int i = 0; i < 16; ++i) Tm[i] = T[i];

    // Loop-invariant A operands: A_k[m][q] = K16[m][4k+q] = T[m/4][k] * T[m%4][q].
    // Lane layout (ISA 16x4 f32 A): lanes 0-15 -> K columns {0,1}, lanes 16-31 -> {2,3}.
    const int L  = lane & 15;
    const int q0 = (lane < 16) ? 0 : 2;
    const float tl0 = T[4 * (L & 3) + q0];        // per-lane vmem load (avoids dyn-indexed regs)
    const float tl1 = T[4 * (L & 3) + q0 + 1];
    v2f Ak[4];
#pragma unroll
    for (int k = 0; k < 4; ++k) {
        const float th = T[4 * (L >> 2) + k];
        Ak[k].x = th * tl0;
        Ak[k].y = th * tl1;
    }

    const float4* cs4 = (const float4*)covs_src;
    const float4* ct4 = (const float4*)covs_tar;
    const float4* sp4 = (const float4*)src;
    const float4* tp4 = (const float4*)tar;

    const int n   = lane & 15;               // sub-point id within a 16-batch
    const int rr0 = (lane < 16) ? 0 : 2;     // row pair this lane loads (B / C-preload layout)

    float acc = 0.0f;

    for (int batch = waveId; batch < numBatches; batch += totalWaves) {
        const int base = batch * 32;
        int p0 = base + n;        // batch0 point, handled by lane pair (n, n+16)
        int p1 = base + 16 + n;   // batch1 point
        const bool valid = (base + lane) < N;
        if (p0 >= N) p0 = N - 1;
        if (p1 >= N) p1 = N - 1;

        // prefetch next tile of the streamed source covariances
        if (batch + totalWaves < numBatches) {
            __builtin_prefetch(&cs4[(size_t)(p0 + totalWaves * 32) * 4 + rr0], 0, 0);
        }

        const int j0 = idx[p0];
        const int j1 = idx[p1];

        // Preload accumulators with gathered C_tar in vec() layout (symmetry: cols == rows).
        float4 g0a = ct4[(size_t)j0 * 4 + rr0];
        float4 g0b = ct4[(size_t)j0 * 4 + rr0 + 1];
        float4 g1a = ct4[(size_t)j1 * 4 + rr0];
        float4 g1b = ct4[(size_t)j1 * 4 + rr0 + 1];
        v8f D0, D1;
        D0[0]=g0a.x; D0[1]=g0a.y; D0[2]=g0a.z; D0[3]=g0a.w;
        D0[4]=g0b.x; D0[5]=g0b.y; D0[6]=g0b.z; D0[7]=g0b.w;
        D1[0]=g1a.x; D1[1]=g1a.y; D1[2]=g1a.z; D1[3]=g1a.w;
        D1[4]=g1b.x; D1[5]=g1b.y; D1[6]=g1b.z; D1[7]=g1b.w;

        // Streamed C_src rows for the B operands (coalesced b128).
        float4 s0a = cs4[(size_t)p0 * 4 + rr0];
        float4 s0b = cs4[(size_t)p0 * 4 + rr0 + 1];
        float4 s1a = cs4[(size_t)p1 * 4 + rr0];
        float4 s1b = cs4[(size_t)p1 * 4 + rr0 + 1];

        // D = C_tar + (T (x) T) vec(C_src)  == vec(RCR), 16 points per chain.
        D0 = kron_sandwich(Ak, s0a, s0b, D0);
        D1 = kron_sandwich(Ak, s1a, s1b, D1);

        // Cross-half exchange: each lane ends up owning one point's symmetric 3x3.
        float x0 = (lane < 16) ? D1[0] : D0[0];
        float x1 = (lane < 16) ? D1[1] : D0[1];
        float x2 = (lane < 16) ? D1[2] : D0[2];
        float x5 = (lane < 16) ? D1[5] : D0[5];
        float r0 = __shfl_xor(x0, 16, 32);
        float r1 = __shfl_xor(x1, 16, 32);
        float r2 = __shfl_xor(x2, 16, 32);
        float r5 = __shfl_xor(x5, 16, 32);

        // Symmetric RCR3 = [[a,b,c],[b,d,e],[c,e,f]]
        float a_, b_, c_, d_, e_, f_;
        if (lane < 16) { a_ = D0[0]; b_ = D0[1]; c_ = D0[2]; d_ = D0[5]; e_ = D0[6]; f_ = r2;   }
        else           { a_ = r0;    b_ = r1;    c_ = D1[0]; d_ = r5;    e_ = D1[1]; f_ = D1[2]; }

        // Residual for this lane's own point (low lanes: p0, high lanes: p1).
        const int pown = (lane < 16) ? p0 : p1;
        const int jown = (lane < 16) ? j0 : j1;
        float4 sp = sp4[pown];
        float4 tp = tp4[(size_t)jown];
        float tx = sp.x * Tm[0] + sp.y * Tm[4] + sp.z * Tm[8]  + sp.w * Tm[12];
        float ty = sp.x * Tm[1] + sp.y * Tm[5] + sp.z * Tm[9]  + sp.w * Tm[13];
        float tz = sp.x * Tm[2] + sp.y * Tm[6] + sp.z * Tm[10] + sp.w * Tm[14];
        float rx = tp.x - tx, ry = tp.y - ty, rz = tp.z - tz;

        // Adjugate-based inverse quadratic form (SPD 3x3).
        float A00 = d_ * f_ - e_ * e_;
        float A01 = c_ * e_ - b_ * f_;
        float A02 = b_ * e_ - c_ * d_;
        float A11 = a_ * f_ - c_ * c_;
        float A12 = b_ * c_ - a_ * e_;
        float A22 = a_ * d_ - b_ * b_;
        float det = a_ * A00 + b_ * A01 + c_ * A02;
        float q = A00 * rx * rx + A11 * ry * ry + A22 * rz * rz
                + 2.0f * (A01 * rx * ry + A02 * rx * rz + A12 * ry * rz);
        acc += valid ? (q / det) : 0.0f;
    }

    // wave32 tree reduce
#pragma unroll
    for (int off = 16; off > 0; off >>= 1) acc += __shfl_xor(acc, off, 32);

    __shared__ float ws[8];
    if (lane == 0) ws[waveInBlock] = acc;
    __syncthreads();
    if (tid == 0) {
        float s = 0.0f;
        for (int w = 0; w < wavesPerBlock; ++w) s += ws[w];
        partial[blockIdx.x] = s;
    }
}

// Deterministic fixed-order final reduction; writes 0.5 * mean.
__global__ __launch_bounds__(256) void gicp_reduce(
    const float* __restrict__ partial, int nParts, float* __restrict__ out, float scale)
{
    __shared__ float s[256];
    float acc = 0.0f;
    for (int i = threadIdx.x; i < nParts; i += 256) acc += partial[i];
    s[threadIdx.x] = acc;
    __syncthreads();
#pragma unroll
    for (int off = 128; off > 0; off >>= 1) {
        if (threadIdx.x < off) s[threadIdx.x] += s[threadIdx.x + off];
        __syncthreads();
    }
    if (threadIdx.x == 0) out[0] = s[0] * scale;
}

extern "C" void kernel_launch(void* const* d_in, const int* in_sizes, int n_in,
                              void* d_out, int out_size, void* d_ws, size_t ws_size,
                              hipStream_t stream) {
    const float* T   = (const float*)d_in[0];  // (4,4)
    const float* src = (const float*)d_in[1];  // (N,4)
    const float* tar = (const float*)d_in[2];  // (M,4)
    const float* cs  = (const float*)d_in[3];  // (N,4,4)
    const float* ct  = (const float*)d_in[4];  // (M,4,4)
    const int*   idx = (const int*)d_in[5];    // (N,)

    const int N = in_sizes[1] / 4;
    const int numBatches = (N + 31) / 32;

    const int threads = 256;
    const int wavesPerBlock = threads / 32;
    int blocks = 2048;
    int maxBlocks = (numBatches + wavesPerBlock - 1) / wavesPerBlock;
    if (blocks > maxBlocks) blocks = maxBlocks;
    if (blocks < 1) blocks = 1;

    float* partial = (float*)d_ws;  // blocks floats of scratch

    gicp_main<<<blocks, threads, 0, stream>>>(T, src, tar, cs, ct, idx, partial, N, numBatches);
    gicp_reduce<<<1, 256, 0, stream>>>(partial, blocks, (float*)d_out, 0.5f / (float)N);
}